// BlackBoxSDE_15590731284541
// MI455X (gfx1250) — compile-verified
//
#include <hip/hip_runtime.h>

// CDNA5 (gfx1250) Euler-Maruyama SDE scan.
// Parallel over batch (B=8192), serial time loop (T=2048) inside the kernel.
// Layer-1 GEMMs done with V_WMMA_F32_16X16X4_F32 (exact fp32), bias fused
// into the unused K=3 slot. Layer-2 reductions done with per-lane FMAs over
// register-resident W2/D2 plus one cross-half shuffle.
// SELU/tanh are branch-free so the 2047-step serial loop has no EXEC churn.

typedef __attribute__((ext_vector_type(2))) float v2f;
typedef __attribute__((ext_vector_type(8))) float v8f;

#define T_STEPS 2048
#define B_SZ    8192
#define H_DIM   128
#define HD_DIM  32
#define BLOCK   256   // 8 wave32 per workgroup
#define COLS_PER_WAVE 16

// selu(x) = lam*max(x,0) + lam*alpha*(exp(min(x,0)) - 1)   (exact identity,
// branch-free: for x>0 the exp term is exp(0)-1 == 0).
__device__ __forceinline__ float selu_f(float x) {
    const float lam       = 1.0507009873554805f;
    const float lam_alpha = 1.7580993408473766f;   // lambda * alpha
    float e = __expf(fminf(x, 0.0f));              // v_exp_f32, always valid
    return fmaf(lam, fmaxf(x, 0.0f), lam_alpha * (e - 1.0f));
}

__device__ __forceinline__ float tanh_f(float x) {
#if __has_builtin(__builtin_amdgcn_tanhf)
    return __builtin_amdgcn_tanhf(x);              // v_tanh_f32
#elif __has_builtin(__builtin_amdgcn_tanh_f32)
    return __builtin_amdgcn_tanh_f32(x);
#else
    // branch-free fallback: tanh(x) = (e^{2x}-1)/(e^{2x}+1), clamped
    float t = fminf(fmaxf(x, -9.0f), 9.0f);
    float e = __expf(2.0f * t);
    return (e - 1.0f) * __builtin_amdgcn_rcpf(e + 1.0f);
#endif
}

__global__ __launch_bounds__(BLOCK) void sde_scan_kernel(
    const float* __restrict__ t,
    const float* __restrict__ u,
    const float* __restrict__ y0in,
    const float* __restrict__ noise,
    const float* __restrict__ W1, const float* __restrict__ b1,
    const float* __restrict__ W2, const float* __restrict__ b2,
    const float* __restrict__ D1, const float* __restrict__ d1,
    const float* __restrict__ D2, const float* __restrict__ d2,
    float* __restrict__ out)
{
    const int lane = threadIdx.x & 31;
    const int wave = (blockIdx.x * BLOCK + threadIdx.x) >> 5;
    const int half = lane >> 4;       // 0: lanes 0-15, 1: lanes 16-31
    const int l16  = lane & 15;
    const int col  = wave * COLS_PER_WAVE + l16;   // batch column this lane serves

    const float dt   = t[1] - t[0];
    const float sqdt = sqrtf(dt);

    // ---- A-matrix tiles for drift layer 1: A[m,k] = W1[k,m], A[m,3] = b1[m] ----
    // 16x4 fp32 A layout: lanes 0-15 hold {K=0,K=1}, lanes 16-31 hold {K=2,K=3}.
    v2f a1[8];
#pragma unroll
    for (int k = 0; k < 8; ++k) {
        int m = k * 16 + l16;
        if (half == 0) { a1[k].x = W1[0*H_DIM + m]; a1[k].y = W1[1*H_DIM + m]; }
        else           { a1[k].x = W1[2*H_DIM + m]; a1[k].y = b1[m];           }
    }
    // ---- A-matrix tiles for diffusion layer 1 (same trick with d1) ----
    v2f ad[2];
#pragma unroll
    for (int k = 0; k < 2; ++k) {
        int m = k * 16 + l16;
        if (half == 0) { ad[k].x = D1[0*HD_DIM + m]; ad[k].y = D1[1*HD_DIM + m]; }
        else           { ad[k].x = D1[2*HD_DIM + m]; ad[k].y = d1[m];            }
    }
    // ---- register-resident reduction weights, sliced per C/D VGPR layout ----
    // C/D 16x16 fp32: VGPR r holds row m = 16*tile + r + 8*half for column l16.
    float w2r[8][8];
#pragma unroll
    for (int k = 0; k < 8; ++k)
#pragma unroll
        for (int r = 0; r < 8; ++r)
            w2r[k][r] = W2[k*16 + r + 8*half];
    float d2a[2][8], d2b[2][8];
#pragma unroll
    for (int k = 0; k < 2; ++k)
#pragma unroll
        for (int r = 0; r < 8; ++r) {
            int m = k*16 + r + 8*half;
            d2a[k][r] = D2[m*2 + 0];
            d2b[k][r] = D2[m*2 + 1];
        }
    const float b2s = b2[0];
    const float d20 = d2[0], d21 = d2[1];

    // ---- state, replicated in both lane halves ----
    float ya = y0in[col*2 + 0];
    float yb = y0in[col*2 + 1];

    if (half == 0) {
        float2 v = make_float2(ya, yb);
        *(float2*)(out + (size_t)col * 2) = v;   // ys[0] = y0
    }

    const v8f cz = {};   // zero accumulator

    for (int s = 0; s < T_STEPS - 1; ++s) {
        // ut[s] == u[s] exactly on this uniform grid (alpha collapses to 0/1)
        const float un = u[s];

        // prefetch the noise stream a few steps ahead (global_prefetch_b8)
        {
            int ps = s + 8; if (ps > T_STEPS - 2) ps = T_STEPS - 2;
            __builtin_prefetch(noise + ((size_t)ps * B_SZ + col) * 2, 0, 0);
        }

        // ---- B matrix (4x16): rows = {y0, y1, u, 1}, cols = batch ----
        // layout mirrors A: lanes 0-15 hold {K=0,K=1}, lanes 16-31 {K=2,K=3}
        v2f bm;
        bm.x = half ? un   : ya;
        bm.y = half ? 1.0f : yb;

        // ---- layer 1: 8 WMMAs (drift, 128 hidden) + 2 WMMAs (diffusion, 32 hidden)
        v8f c[8];
#pragma unroll
        for (int k = 0; k < 8; ++k)
            c[k] = __builtin_amdgcn_wmma_f32_16x16x4_f32(
                false, a1[k], false, bm, (short)0, cz, false, false);
        v8f cd0 = __builtin_amdgcn_wmma_f32_16x16x4_f32(
                false, ad[0], false, bm, (short)0, cz, false, false);
        v8f cd1 = __builtin_amdgcn_wmma_f32_16x16x4_f32(
                false, ad[1], false, bm, (short)0, cz, false, false);

        // ---- drift: SELU + dot with W2 (per-lane partial over 64 rows) ----
        float acc = 0.0f;
#pragma unroll
        for (int k = 0; k < 8; ++k)
#pragma unroll
            for (int r = 0; r < 8; ++r)
                acc = fmaf(selu_f(c[k][r]), w2r[k][r], acc);

        // ---- diffusion: tanh + dot with D2 columns ----
        float g0p = 0.0f, g1p = 0.0f;
#pragma unroll
        for (int r = 0; r < 8; ++r) {
            float h0 = tanh_f(cd0[r]);
            float h1 = tanh_f(cd1[r]);
            g0p = fmaf(h0, d2a[0][r], g0p); g1p = fmaf(h0, d2b[0][r], g1p);
            g0p = fmaf(h1, d2a[1][r], g0p); g1p = fmaf(h1, d2b[1][r], g1p);
        }

        // fold the two half-wave row groups (lane n <-> lane n+16)
        acc += __shfl_xor(acc, 16, 32);
        g0p += __shfl_xor(g0p, 16, 32);
        g1p += __shfl_xor(g1p, 16, 32);

        const float thdd = acc + b2s;
        const float g0   = g0p + d20;
        const float g1   = g1p + d21;

        // ---- Euler-Maruyama update (both halves compute identically) ----
        float2 z = *(const float2*)(noise + ((size_t)s * B_SZ + col) * 2);
        float yna = fmaf(yb,   dt, fmaf(g0 * sqdt, z.x, ya));
        float ynb = fmaf(thdd, dt, fmaf(g1 * sqdt, z.y, yb));
        ya = yna; yb = ynb;

        if (half == 0) {
            float2 v = make_float2(ya, yb);
            *(float2*)(out + ((size_t)(s + 1) * B_SZ + col) * 2) = v;
        }
    }
}

extern "C" void kernel_launch(void* const* d_in, const int* in_sizes, int n_in,
                              void* d_out, int out_size, void* d_ws, size_t ws_size,
                              hipStream_t stream) {
    const float* t     = (const float*)d_in[0];
    const float* u     = (const float*)d_in[1];
    const float* y0    = (const float*)d_in[2];
    const float* noise = (const float*)d_in[3];
    const float* W1    = (const float*)d_in[4];
    const float* b1    = (const float*)d_in[5];
    const float* W2    = (const float*)d_in[6];
    const float* b2    = (const float*)d_in[7];
    const float* D1    = (const float*)d_in[8];
    const float* d1    = (const float*)d_in[9];
    const float* D2    = (const float*)d_in[10];
    const float* d2    = (const float*)d_in[11];
    float* out = (float*)d_out;

    const int waves   = B_SZ / COLS_PER_WAVE;        // 512
    const int threads = waves * 32;                  // 16384
    const int grid    = threads / BLOCK;             // 64 blocks of 8 waves

    sde_scan_kernel<<<grid, BLOCK, 0, stream>>>(
        t, u, y0, noise, W1, b1, W2, b2, D1, d1, D2, d2, out);
}